// CausalSelfAttention_7335804141636
// MI455X (gfx1250) — compile-verified
//
#include <hip/hip_runtime.h>
#include <math.h>

// Problem constants (from reference)
#define BATCH  16
#define CDIM   2048
#define NH     16
#define HS     128
#define TPAST  4095
#define TTOT   4096
#define QKVN   (3 * CDIM)   // 6144
#define CHUNK  512
#define NCHUNK 8            // 8*512 = 4096 scores per (b,h); last slot = fresh token
#define PSTRIDE (HS + 2)    // per-partial record: acc[128], m, l

typedef float v2f __attribute__((ext_vector_type(2)));
typedef float v8f __attribute__((ext_vector_type(8)));

// ---------------------------------------------------------------------------
// GEMM: Out[16,N] = X[16,K] @ W[K,N] + bias[N], fp32 WMMA 16x16x4.
// One wave per 16-column tile; M=16 == batch; full fp32 precision.
// A-frag (ISA 7.12.2): lanes 0-15 hold K=k,k+1 for M=lane; lanes 16-31 K=k+2,k+3.
// ---------------------------------------------------------------------------
__global__ void __launch_bounds__(32)
gemm16_wmma(const float* __restrict__ X, const float* __restrict__ W,
            const float* __restrict__ bias, float* __restrict__ Out,
            int N, int K) {
  const int lane = threadIdx.x & 31;
  const int half = lane >> 4;
  const int m    = lane & 15;
  const int col  = blockIdx.x * 16 + m;

  v8f acc = {0.f, 0.f, 0.f, 0.f, 0.f, 0.f, 0.f, 0.f};
  const float* xrow = X + (size_t)m * K;

#pragma unroll 4
  for (int kk = 0; kk < K; kk += 4) {
    const int k0 = kk + half * 2;
    v2f a, b;
    a.x = xrow[k0];
    a.y = xrow[k0 + 1];
    const float* wp = W + (size_t)k0 * N + col;
    b.x = wp[0];
    b.y = wp[N];
    acc = __builtin_amdgcn_wmma_f32_16x16x4_f32(
        false, a, false, b, (short)0, acc, false, false);
  }

  const float bv = bias[col];
#pragma unroll
  for (int r = 0; r < 8; ++r) {
    const int row = r + half * 8;
    Out[(size_t)row * N + col] = acc[r] + bv;
  }
}

// ---------------------------------------------------------------------------
// Fused split-KV attention chunk + cache re-quantization.
// Block = 128 threads (4 waves); blockIdx.x = bh*NCHUNK + c; chunk c covers
// tokens [c*512, c*512+512) (last chunk: 511 past rows + the fresh token).
//
// Phase A: wave-per-K-row: q·k dot AND max|k| via one dual shuffle tree;
//          emits k_quant/k_scale; scaled score -> LDS.
// Phase B: chunk-local online-softmax (m, l); weight w = e^(s-m) * v_scale
//          folded into LDS (attention uses the DEQUANTIZED V, per reference).
// Phase C: wave-per-V-row: emits v_quant/v_scale and accumulates w*V into
//          per-wave y partials; waves combined through LDS.
// Writes partial record {acc[128], m, l} for the combine kernel.
// ---------------------------------------------------------------------------
__global__ void __launch_bounds__(128)
attn_chunk(const float* __restrict__ qkv,
           const int* __restrict__ pkq, const float* __restrict__ pks,
           const int* __restrict__ pvq, const float* __restrict__ pvs,
           float* __restrict__ kq_out, float* __restrict__ ks_out,
           float* __restrict__ vq_out, float* __restrict__ vs_out,
           float* __restrict__ part) {
  __shared__ float s_p[CHUNK];
  __shared__ float s_red[128];
  __shared__ float s_y[4][HS];

  const int tid  = threadIdx.x;
  const int lane = tid & 31;
  const int wave = tid >> 5;
  const int blk  = blockIdx.x;
  const int c    = blk & (NCHUNK - 1);
  const int bh   = blk >> 3;
  const int b    = bh >> 4;
  const int h    = bh & 15;
  const int t0   = c * CHUNK;
  const bool last = (c == NCHUNK - 1);
  const int npast = last ? (CHUNK - 1) : CHUNK;   // past rows in this chunk
  const float rs = 0.08838834764831845f;          // 1/sqrt(128)

  // q fragment: lane owns dims [lane*4, lane*4+4)
  const float4 q4 =
      *(const float4*)(qkv + (size_t)b * QKVN + h * HS + lane * 4);

  const int*   Kb  = pkq + ((size_t)bh * TPAST + t0) * HS;
  const float* Ks  = pks + (size_t)bh * TPAST + t0;
  const float* Vs  = pvs + (size_t)bh * TPAST + t0;
  float* kqo = kq_out + ((size_t)bh * TTOT + t0) * HS;
  float* kso = ks_out + (size_t)bh * TTOT + t0;

  // ---- Phase A: K stream -> scores + k re-quant ----
  for (int i = wave; i < npast; i += 4) {
    const int* kp = Kb + (size_t)i * HS + lane * 4;
    __builtin_prefetch(kp + 16 * HS, 0, 0);       // global_prefetch_b8
    const int4 kq = *(const int4*)kp;
    const float f0 = (float)kq.x, f1 = (float)kq.y;
    const float f2 = (float)kq.z, f3 = (float)kq.w;
    float dot = q4.x * f0 + q4.y * f1 + q4.z * f2 + q4.w * f3;
    float mxa = fmaxf(fmaxf(fabsf(f0), fabsf(f1)), fmaxf(fabsf(f2), fabsf(f3)));
#pragma unroll
    for (int off = 16; off; off >>= 1) {          // dual reduction: sum + max
      dot += __shfl_xor(dot, off, 32);
      mxa = fmaxf(mxa, __shfl_xor(mxa, off, 32));
    }
    const float inv = (mxa > 0.f) ? 127.f / mxa : 0.f;
    float4 qv;
    qv.x = rintf(f0 * inv); qv.y = rintf(f1 * inv);
    qv.z = rintf(f2 * inv); qv.w = rintf(f3 * inv);
    *(float4*)(kqo + (size_t)i * HS + lane * 4) = qv;
    if (lane == 0) {
      const float s = Ks[i];
      kso[i] = s * mxa * (1.f / 127.f);
      s_p[i] = dot * s * rs;
    }
  }
  if (last && wave == 0) {  // fresh token's k (fp32, from qkv buffer)
    const float4 kn =
        *(const float4*)(qkv + (size_t)b * QKVN + CDIM + h * HS + lane * 4);
    float dot = q4.x * kn.x + q4.y * kn.y + q4.z * kn.z + q4.w * kn.w;
    float mxa = fmaxf(fmaxf(fabsf(kn.x), fabsf(kn.y)),
                      fmaxf(fabsf(kn.z), fabsf(kn.w)));
#pragma unroll
    for (int off = 16; off; off >>= 1) {
      dot += __shfl_xor(dot, off, 32);
      mxa = fmaxf(mxa, __shfl_xor(mxa, off, 32));
    }
    const float inv = (mxa > 0.f) ? 127.f / mxa : 0.f;
    float4 qv;
    qv.x = rintf(kn.x * inv); qv.y = rintf(kn.y * inv);
    qv.z = rintf(kn.z * inv); qv.w = rintf(kn.w * inv);
    *(float4*)(kqo + (size_t)(CHUNK - 1) * HS + lane * 4) = qv;
    if (lane == 0) {
      kso[CHUNK - 1] = mxa * (1.f / 127.f);
      s_p[CHUNK - 1] = dot * rs;
    }
  }
  __syncthreads();

  // ---- Phase B: chunk-local softmax; fold v_scale into weights ----
  float mx = -3.4e38f;
  for (int i = tid; i < CHUNK; i += 128) mx = fmaxf(mx, s_p[i]);
  s_red[tid] = mx;
  __syncthreads();
  for (int s = 64; s > 0; s >>= 1) {
    if (tid < s) s_red[tid] = fmaxf(s_red[tid], s_red[tid + s]);
    __syncthreads();
  }
  const float m_local = s_red[0];
  __syncthreads();

  float lsum = 0.f;
  for (int i = tid; i < CHUNK; i += 128) {
    const float e = __expf(s_p[i] - m_local);
    lsum += e;
    s_p[i] = (i < npast) ? e * Vs[i] : e;   // dequant scale of past V rows
  }
  s_red[tid] = lsum;
  __syncthreads();
  for (int s = 64; s > 0; s >>= 1) {
    if (tid < s) s_red[tid] += s_red[tid + s];
    __syncthreads();
  }
  const float l_local = s_red[0];

  // ---- Phase C: V stream -> y partials + v re-quant ----
  const int* Vb = pvq + ((size_t)bh * TPAST + t0) * HS;
  float* vqo = vq_out + ((size_t)bh * TTOT + t0) * HS;
  float* vso = vs_out + (size_t)bh * TTOT + t0;
  float yacc[4] = {0.f, 0.f, 0.f, 0.f};

  for (int i = wave; i < npast; i += 4) {
    const int* vp = Vb + (size_t)i * HS + lane * 4;
    __builtin_prefetch(vp + 16 * HS, 0, 0);
    const int4 vq = *(const int4*)vp;
    const float f0 = (float)vq.x, f1 = (float)vq.y;
    const float f2 = (float)vq.z, f3 = (float)vq.w;
    float mxa = fmaxf(fmaxf(fabsf(f0), fabsf(f1)), fmaxf(fabsf(f2), fabsf(f3)));
#pragma unroll
    for (int off = 16; off; off >>= 1) mxa = fmaxf(mxa, __shfl_xor(mxa, off, 32));
    const float inv = (mxa > 0.f) ? 127.f / mxa : 0.f;
    float4 qv;
    qv.x = rintf(f0 * inv); qv.y = rintf(f1 * inv);
    qv.z = rintf(f2 * inv); qv.w = rintf(f3 * inv);
    *(float4*)(vqo + (size_t)i * HS + lane * 4) = qv;
    if (lane == 0) vso[i] = Vs[i] * mxa * (1.f / 127.f);
    const float w = s_p[i];                 // e * v_scale already folded
    yacc[0] += w * f0; yacc[1] += w * f1;
    yacc[2] += w * f2; yacc[3] += w * f3;
  }
  if (last && wave == 0) {  // fresh token's v
    const float4 vn =
        *(const float4*)(qkv + (size_t)b * QKVN + 2 * CDIM + h * HS + lane * 4);
    float mxa = fmaxf(fmaxf(fabsf(vn.x), fabsf(vn.y)),
                      fmaxf(fabsf(vn.z), fabsf(vn.w)));
#pragma unroll
    for (int off = 16; off; off >>= 1) mxa = fmaxf(mxa, __shfl_xor(mxa, off, 32));
    const float inv = (mxa > 0.f) ? 127.f / mxa : 0.f;
    float4 qv;
    qv.x = rintf(vn.x * inv); qv.y = rintf(vn.y * inv);
    qv.z = rintf(vn.z * inv); qv.w = rintf(vn.w * inv);
    *(float4*)(vqo + (size_t)(CHUNK - 1) * HS + lane * 4) = qv;
    if (lane == 0) vso[CHUNK - 1] = mxa * (1.f / 127.f);
    const float w = s_p[CHUNK - 1];
    yacc[0] += w * vn.x; yacc[1] += w * vn.y;
    yacc[2] += w * vn.z; yacc[3] += w * vn.w;
  }

  // combine the 4 wave partials
#pragma unroll
  for (int j = 0; j < 4; ++j) s_y[wave][lane * 4 + j] = yacc[j];
  __syncthreads();

  float* P = part + (size_t)blk * PSTRIDE;
  P[tid] = s_y[0][tid] + s_y[1][tid] + s_y[2][tid] + s_y[3][tid];
  if (tid == 0) { P[HS] = m_local; P[HS + 1] = l_local; }
}

// ---------------------------------------------------------------------------
// Combine NCHUNK online-softmax partials per (b,h):
//   y[d] = sum_c e^(m_c - m) * acc_c[d] / sum_c e^(m_c - m) * l_c
// ---------------------------------------------------------------------------
__global__ void __launch_bounds__(128)
attn_combine(const float* __restrict__ part, float* __restrict__ yatt) {
  const int tid = threadIdx.x;       // head dim
  const int bh  = blockIdx.x;
  const int b   = bh >> 4;
  const int h   = bh & 15;
  const float* P0 = part + (size_t)bh * NCHUNK * PSTRIDE;

  float mc[NCHUNK];
  float mg = -3.4e38f;
#pragma unroll
  for (int c = 0; c < NCHUNK; ++c) {
    mc[c] = P0[(size_t)c * PSTRIDE + HS];     // broadcast load
    mg = fmaxf(mg, mc[c]);
  }
  float L = 0.f, acc = 0.f;
#pragma unroll
  for (int c = 0; c < NCHUNK; ++c) {
    const float w = __expf(mc[c] - mg);
    L   += w * P0[(size_t)c * PSTRIDE + HS + 1];
    acc += w * P0[(size_t)c * PSTRIDE + tid];
  }
  yatt[(size_t)b * CDIM + h * HS + tid] = acc / L;
}

// ---------------------------------------------------------------------------
// Launcher
// ---------------------------------------------------------------------------
extern "C" void kernel_launch(void* const* d_in, const int* in_sizes, int n_in,
                              void* d_out, int out_size, void* d_ws,
                              size_t ws_size, hipStream_t stream) {
  (void)in_sizes; (void)n_in; (void)out_size; (void)ws_size;

  const float* x      = (const float*)d_in[0];
  const float* W_attn = (const float*)d_in[1];
  const float* b_attn = (const float*)d_in[2];
  const float* W_proj = (const float*)d_in[3];
  const float* b_proj = (const float*)d_in[4];
  const int*   pkq    = (const int*)d_in[5];
  const float* pks    = (const float*)d_in[6];
  const int*   pvq    = (const int*)d_in[7];
  const float* pvs    = (const float*)d_in[8];

  float* out  = (float*)d_out;
  float* qkv  = (float*)d_ws;                       // [16, 6144]
  float* yatt = qkv + (size_t)BATCH * QKVN;         // [16, 2048]
  float* part = yatt + (size_t)BATCH * CDIM;        // [256, 8, 130]

  // Output layout: y | k_quant | k_scale | v_quant | v_scale (flat, float)
  float* y_out  = out;
  float* kq_out = out + (size_t)BATCH * CDIM;
  float* ks_out = kq_out + (size_t)BATCH * NH * TTOT * HS;
  float* vq_out = ks_out + (size_t)BATCH * NH * TTOT;
  float* vs_out = vq_out + (size_t)BATCH * NH * TTOT * HS;

  // 1) qkv = x @ W_attn + b_attn   (fp32 WMMA, 384 column tiles)
  gemm16_wmma<<<QKVN / 16, 32, 0, stream>>>(x, W_attn, b_attn, qkv, QKVN, CDIM);

  // 2) fused split-KV attention + cache re-quant: 2048 blocks (8 chunks x 256 bh)
  attn_chunk<<<BATCH * NH * NCHUNK, 128, 0, stream>>>(
      qkv, pkq, pks, pvq, pvs, kq_out, ks_out, vq_out, vs_out, part);

  // 3) combine partials: one block per (b,h)
  attn_combine<<<BATCH * NH, 128, 0, stream>>>(part, yatt);

  // 4) y = yatt @ W_proj + b_proj  (fp32 WMMA, 128 column tiles)
  gemm16_wmma<<<CDIM / 16, 32, 0, stream>>>(yatt, W_proj, b_proj, y_out,
                                            CDIM, CDIM);
}